// TwoLayerGCN_10204842295478
// MI455X (gfx1250) — compile-verified
//
#include <hip/hip_runtime.h>
#include <hip/hip_bf16.h>

typedef __attribute__((ext_vector_type(2))) float v2f;
typedef __attribute__((ext_vector_type(8))) float v8f;

// ---------------------------------------------------------------------------
// Degree: deg[dst[e]] += 1 over the 800k real edges (self loops folded later)
// ---------------------------------------------------------------------------
__global__ void gcn_degree(const int* __restrict__ dst, float* __restrict__ deg, int E) {
    int e = blockIdx.x * blockDim.x + threadIdx.x;
    if (e < E) atomicAdd(&deg[dst[e]], 1.0f);
}

// dinv[i] = rsqrt(deg[i] + 1)   (self loop adds 1 to every node; deg >= 1 > 0)
__global__ void gcn_dinv(float* __restrict__ deg_dinv, int Nn) {
    int i = blockIdx.x * blockDim.x + threadIdx.x;
    if (i < Nn) deg_dinv[i] = rsqrtf(deg_dinv[i] + 1.0f);
}

// norm[e] = dinv[src[e]] * dinv[dst[e]]; e in [E, E+Nn) are self loops
__global__ void gcn_norm(const int* __restrict__ src, const int* __restrict__ dst,
                         const float* __restrict__ dinv, float* __restrict__ norm,
                         int E, int Nn) {
    int e = blockIdx.x * blockDim.x + threadIdx.x;
    if (e >= E + Nn) return;
    int s, d;
    if (e < E) { s = src[e]; d = dst[e]; }
    else       { s = e - E; d = s; }
    norm[e] = dinv[s] * dinv[d];
}

// ---------------------------------------------------------------------------
// FP32 WMMA GEMM: C[M x Nout] = A[M x K] * B[K x Nout]
// One wave = one 16x16 output tile; blockDim.x = 32 * (Nout/16); grid = M/16.
// Uses V_WMMA_F32_16X16X4_F32 (A: 16x4 f32 = 2 VGPR, B: 4x16 f32 = 2 VGPR,
// C/D: 8 VGPR). Lane decode per CDNA5 ISA 7.12.2.
// ---------------------------------------------------------------------------
__global__ void gcn_gemm_wmma(const float* __restrict__ A, const float* __restrict__ B,
                              float* __restrict__ C, int M, int K, int Nout) {
    const int wave = threadIdx.x >> 5;
    const int lane = threadIdx.x & 31;
    const int n0   = wave * 16;
    const int m0   = blockIdx.x * 16;
    const int mr   = lane & 15;   // A: row within tile / B,C: column within tile
    const int hi   = lane >> 4;   // lane-half selects K pair (A/B) or M+8 (C/D)
    const int kb   = hi * 2;

    int arow = m0 + mr;
    if (arow >= M) arow = M - 1;              // clamp (keeps EXEC all-ones)
    const float* Arow = A + (long long)arow * K;

    v8f c = {};
    for (int k0 = 0; k0 < K; k0 += 4) {
        v2f a;
        a.x = Arow[k0 + kb];
        a.y = Arow[k0 + kb + 1];
        v2f b;
        b.x = B[(k0 + kb)     * Nout + n0 + mr];
        b.y = B[(k0 + kb + 1) * Nout + n0 + mr];
        // 8 args: (neg_a, A, neg_b, B, c_mod, C, reuse_a, reuse_b)
        c = __builtin_amdgcn_wmma_f32_16x16x4_f32(false, a, false, b,
                                                  (short)0, c, false, false);
    }
#pragma unroll
    for (int r = 0; r < 8; ++r) {
        int row = m0 + r + 8 * hi;
        if (row < M) C[(long long)row * Nout + n0 + mr] = c[r];
    }
}

// ---------------------------------------------------------------------------
// Edge-parallel scatter-add: out[dst[e], f] += h[src[e], f] * norm[e]
// Feature index lives in the lane dimension -> coalesced gathers and
// 32-consecutive-word atomic bursts (L2-resident: h fits in 192MB L2).
// F is a power of two; shift = log2(F).
// ---------------------------------------------------------------------------
__global__ void gcn_aggregate(const int* __restrict__ src, const int* __restrict__ dst,
                              const float* __restrict__ norm, const float* __restrict__ h,
                              float* __restrict__ out, int E, int Nn, int F, int shift) {
    long long tid = (long long)blockIdx.x * blockDim.x + threadIdx.x;
    int e = (int)(tid >> shift);
    int f = (int)(tid & (long long)(F - 1));
    if (e >= E + Nn) return;
    int s, d;
    if (e < E) { s = src[e]; d = dst[e]; }
    else       { s = e - E; d = s; }
    float v = h[(long long)s * F + f] * norm[e];
    atomicAdd(out + (long long)d * F + f, v);   // global_atomic_add_f32 (no return)
}

// out[i] = relu(in[i] + bias[i & (F-1)])   (in may alias out)
__global__ void gcn_bias_relu(const float* in, const float* __restrict__ bias,
                              float* out, long long total, int F) {
    long long i = (long long)blockIdx.x * blockDim.x + threadIdx.x;
    if (i >= total) return;
    float v = in[i] + bias[(int)(i & (long long)(F - 1))];
    out[i] = v > 0.0f ? v : 0.0f;
}

// ---------------------------------------------------------------------------
extern "C" void kernel_launch(void* const* d_in, const int* in_sizes, int n_in,
                              void* d_out, int out_size, void* d_ws, size_t ws_size,
                              hipStream_t stream) {
    const float* x  = (const float*)d_in[0];   // [N, 64]
    const int*   ei = (const int*)  d_in[1];   // [2, E]
    const float* W1 = (const float*)d_in[2];   // [64, 128]
    const float* b1 = (const float*)d_in[3];   // [128]
    const float* W2 = (const float*)d_in[4];   // [128, 64]
    const float* b2 = (const float*)d_in[5];   // [64]
    float* out = (float*)d_out;                // [N, 64]

    const int IN = 64, HID = 128, OUT = 64;
    const int Nn = in_sizes[0] / IN;           // 50000
    const int E  = in_sizes[1] / 2;            // 800000
    const int Etot = E + Nn;                   // edges + self loops
    const int* src = ei;
    const int* dst = ei + E;

    // workspace layout (256B aligned slices)
    auto rnd = [](size_t b) { return (b + 255) & ~(size_t)255; };
    char* ws = (char*)d_ws;
    float* dinv = (float*)ws;  ws += rnd((size_t)Nn * 4);
    float* nrm  = (float*)ws;  ws += rnd((size_t)Etot * 4);
    float* h1   = (float*)ws;  ws += rnd((size_t)Nn * HID * 4);   // also reused for a1
    float* agg1 = (float*)ws;  ws += rnd((size_t)Nn * HID * 4);
    float* h2   = (float*)ws;  ws += rnd((size_t)Nn * OUT * 4);

    // zero accumulators (memset nodes are graph-capturable)
    hipMemsetAsync(dinv, 0, (size_t)Nn * 4, stream);
    hipMemsetAsync(agg1, 0, (size_t)Nn * HID * 4, stream);
    hipMemsetAsync(out,  0, (size_t)out_size * 4, stream);

    // normalization
    gcn_degree<<<(E + 255) / 256, 256, 0, stream>>>(dst, dinv, E);
    gcn_dinv  <<<(Nn + 255) / 256, 256, 0, stream>>>(dinv, Nn);
    gcn_norm  <<<(Etot + 255) / 256, 256, 0, stream>>>(src, dst, dinv, nrm, E, Nn);

    // layer 1: h1 = x @ W1 ; agg1 = scatter(h1*norm) ; a1 = relu(agg1 + b1)
    gcn_gemm_wmma<<<(Nn + 15) / 16, 32 * (HID / 16), 0, stream>>>(x, W1, h1, Nn, IN, HID);
    {
        long long t = (long long)Etot * HID;
        gcn_aggregate<<<(unsigned)((t + 255) / 256), 256, 0, stream>>>(
            src, dst, nrm, h1, agg1, E, Nn, HID, 7);
        long long tot = (long long)Nn * HID;
        gcn_bias_relu<<<(unsigned)((tot + 255) / 256), 256, 0, stream>>>(
            agg1, b1, h1, tot, HID);                 // a1 overwrites h1 buffer
    }

    // layer 2: h2 = a1 @ W2 ; out = relu(scatter(h2*norm) + b2)
    gcn_gemm_wmma<<<(Nn + 15) / 16, 32 * (OUT / 16), 0, stream>>>(h1, W2, h2, Nn, HID, OUT);
    {
        long long t = (long long)Etot * OUT;
        gcn_aggregate<<<(unsigned)((t + 255) / 256), 256, 0, stream>>>(
            src, dst, nrm, h2, out, E, Nn, OUT, 6);
        long long tot = (long long)Nn * OUT;
        gcn_bias_relu<<<(unsigned)((tot + 255) / 256), 256, 0, stream>>>(
            out, b2, out, tot, OUT);
    }
}